// KREmbedding_39934605918673
// MI455X (gfx1250) — compile-verified
//
#include <hip/hip_runtime.h>

#define VDIM 512
#define CWIN 32
#define ROWP 528   // padded LDS row stride (floats): shifts banks by 16 between c and c+1

typedef __attribute__((ext_vector_type(2))) float v2f;
typedef __attribute__((ext_vector_type(8))) float v8f;
typedef __attribute__((ext_vector_type(4))) int   v4i;

#define GAS __attribute__((address_space(1)))
#define LAS __attribute__((address_space(3)))

#if defined(__has_builtin)
#  if __has_builtin(__builtin_amdgcn_global_load_async_to_lds_b128)
#    define HAVE_ASYNC_LDS 1
#  endif
#endif

__global__ __launch_bounds__(256)
void KREmbedding_39934605918673_kernel(const int* __restrict__ ctx_idx,
                                       const int* __restrict__ cen_idx,
                                       const float* __restrict__ W,
                                       float* __restrict__ out)
{
    __shared__ __align__(16) float s_ctx[CWIN * ROWP];  // 66 KB, padded rows
    __shared__ __align__(16) float s_cen[VDIM];         // 2 KB
    __shared__ __align__(16) float s_out[VDIM];         // 2 KB
    __shared__ float s_dist[CWIN];
    __shared__ float s_w[CWIN];

    const int b    = blockIdx.x;
    const int tid  = threadIdx.x;
    const int lane = tid & 31;
    const int wave = tid >> 5;

    // ---- stage center row into LDS (128 x b128 loads) ----
    {
        const float4* src = (const float4*)(W + (size_t)cen_idx[b] * VDIM);
        float4* dst = (float4*)s_cen;
        for (int i = tid; i < VDIM / 4; i += 256) dst[i] = src[i];
    }

    // ---- stage 32 gathered context rows into LDS (each wave: 4 rows) ----
    for (int r = wave; r < CWIN; r += 8) {
        const float4* src = (const float4*)(W + (size_t)ctx_idx[b * CWIN + r] * VDIM);
        float4* dst = (float4*)(s_ctx + r * ROWP);
        for (int i = lane; i < VDIM / 4; i += 32) {
#if defined(HAVE_ASYNC_LDS)
            __builtin_amdgcn_global_load_async_to_lds_b128(
                (GAS v4i*)(void*)(src + i),
                (LAS v4i*)(void*)(dst + i),
                /*offset=*/0, /*cpol=*/0);
#else
            dst[i] = src[i];
#endif
        }
    }
#if defined(HAVE_ASYNC_LDS)
    asm volatile("s_wait_asynccnt 0" ::: "memory");
#endif
    __syncthreads();

    // ---- squared distances: one wave per 4 context rows ----
    for (int r = wave; r < CWIN; r += 8) {
        const float* row = s_ctx + r * ROWP;
        float acc = 0.0f;
#pragma unroll
        for (int k = 0; k < VDIM / 32; ++k) {
            const int d = lane + k * 32;
            const float df = row[d] - s_cen[d];
            acc = fmaf(df, df, acc);
        }
#pragma unroll
        for (int off = 16; off > 0; off >>= 1)
            acc += __shfl_xor(acc, off, 32);
        if (lane == 0) s_dist[r] = acc;
    }
    __syncthreads();

    // ---- weights (wave 0): w = exp(-d/2); w /= (sum(w) + 1e-8)  [matches reference] ----
    if (wave == 0) {
        const float wv = expf(-0.5f * s_dist[lane]);
        float sum = wv;
#pragma unroll
        for (int off = 16; off > 0; off >>= 1)
            sum += __shfl_xor(sum, off, 32);
        s_w[lane] = wv / (sum + 1e-8f);
    }
    __syncthreads();

    // ---- weighted sum via V_WMMA_F32_16X16X4_F32 ----
    // D[m,n] = sum_k A[m,k] * B[k,n]; A[m,k] = ctx[c0+k][d0+m], B[k,n] = w[c0+k] (bcast over n)
    // => every column of D equals out[d0 .. d0+15]; accumulate over c0 chunks of 4.
    // A layout (ISA 7.12.2, 32-bit A 16x4): lanes 0-15 hold K=0,1; lanes 16-31 hold K=2,3.
    // B layout (4x16, 2 VGPRs): v0 = rows K=0 (lanes 0-15) / K=1 (lanes 16-31); v1 = K=2 / K=3.
    const int half = lane >> 4;   // 0: lanes 0-15, 1: lanes 16-31
    const int ml   = lane & 15;
    for (int t = wave; t < VDIM / 16; t += 8) {  // 32 d-tiles, 4 per wave
        const int d0 = t * 16;
        v8f acc = {0.f, 0.f, 0.f, 0.f, 0.f, 0.f, 0.f, 0.f};
#pragma unroll
        for (int c0 = 0; c0 < CWIN; c0 += 4) {
            v2f a, bm;
            a[0]  = s_ctx[(c0 + 2 * half + 0) * ROWP + d0 + ml];
            a[1]  = s_ctx[(c0 + 2 * half + 1) * ROWP + d0 + ml];
            bm[0] = s_w[c0 + half];
            bm[1] = s_w[c0 + 2 + half];
            acc = __builtin_amdgcn_wmma_f32_16x16x4_f32(
                false, a, false, bm, (short)0, acc, false, false);
        }
        // D layout: VGPR r holds M=r (lanes 0-15) and M=8+r (lanes 16-31); all N identical.
        // Lane 0 carries out[d0..d0+7], lane 16 carries out[d0+8..d0+15].
        if (ml == 0) {
#pragma unroll
            for (int i = 0; i < 8; ++i) s_out[d0 + half * 8 + i] = acc[i];
        }
    }
    __syncthreads();

    // ---- coalesced store of this batch's 512-float output row ----
    {
        float* orow = out + (size_t)b * VDIM;
        for (int i = tid; i < VDIM; i += 256) orow[i] = s_out[i];
    }
}

extern "C" void kernel_launch(void* const* d_in, const int* in_sizes, int n_in,
                              void* d_out, int out_size, void* d_ws, size_t ws_size,
                              hipStream_t stream) {
    (void)n_in; (void)d_ws; (void)ws_size; (void)out_size;
    const int*   context = (const int*)d_in[0];   // [B, C] int32
    const int*   center  = (const int*)d_in[1];   // [B] int32
    const float* W       = (const float*)d_in[2]; // [V, D] f32
    float*       out     = (float*)d_out;         // [B, D] f32
    const int B = in_sizes[1];                    // 8192 batches
    KREmbedding_39934605918673_kernel<<<B, 256, 0, stream>>>(context, center, W, out);
}